// NodesToEdges_21088289424018
// MI455X (gfx1250) — compile-verified
//
#include <hip/hip_runtime.h>
#include <math.h>
#include <stdint.h>

#define DIM_NODES 128
#define DIM_EDGES 64
#define BATCH 2
#define NPTS 512
#define EPS_F32 1.1920929e-7f   // jnp.finfo(float32).eps

typedef __attribute__((ext_vector_type(2))) float v2f;
typedef __attribute__((ext_vector_type(8))) float v8f;

// ---------------------------------------------------------------------------
// CDNA5 async global->LDS support (ASYNCcnt-tracked DMA, no VGPR round trip)
// Builtin signature (from clang diagnostic): (v4i as1*, v4i as3*, imm, imm)
// ---------------------------------------------------------------------------
#if defined(__gfx1250__) && __has_builtin(__builtin_amdgcn_global_load_async_to_lds_b128)
#define USE_ASYNC_LDS 1
#else
#define USE_ASYNC_LDS 0
#endif

#if USE_ASYNC_LDS
typedef int v4i __attribute__((vector_size(16)));
typedef __attribute__((address_space(1))) v4i* gv4i_p;   // global (AS1) v4i*
typedef __attribute__((address_space(3))) v4i* lv4i_p;   // LDS    (AS3) v4i*

__device__ __forceinline__ gv4i_p to_glb(const void* p) {
    return (gv4i_p)(uintptr_t)p;                    // flat addr == global addr
}
__device__ __forceinline__ lv4i_p to_lds(const void* p) {
    return (lv4i_p)(uint32_t)(uintptr_t)p;          // flat low 32b == LDS offset
}
#if __has_builtin(__builtin_amdgcn_s_wait_asynccnt)
#define WAIT_ASYNC(n) __builtin_amdgcn_s_wait_asynccnt(n)
#else
#define WAIT_ASYNC(n) asm volatile("s_wait_asynccnt %0" ::"i"(n) : "memory")
#endif
// one 16x64 f32 tile = 16 rows; lane -> (row = lane>>1, 32-float half-row);
// 8 x b128 per lane with immediate byte offsets 0..112.
#define ASYNC_TILE_LOAD(g, l)                                            \
    do {                                                                 \
        __builtin_amdgcn_global_load_async_to_lds_b128((g), (l), 0, 0);  \
        __builtin_amdgcn_global_load_async_to_lds_b128((g), (l), 16, 0); \
        __builtin_amdgcn_global_load_async_to_lds_b128((g), (l), 32, 0); \
        __builtin_amdgcn_global_load_async_to_lds_b128((g), (l), 48, 0); \
        __builtin_amdgcn_global_load_async_to_lds_b128((g), (l), 64, 0); \
        __builtin_amdgcn_global_load_async_to_lds_b128((g), (l), 80, 0); \
        __builtin_amdgcn_global_load_async_to_lds_b128((g), (l), 96, 0); \
        __builtin_amdgcn_global_load_async_to_lds_b128((g), (l), 112, 0);\
    } while (0)
#endif

// ---------------------------------------------------------------------------
// Kernel 1: per (b,i) node RMS-norm + 128->64 projections.
//   rowcb[bi*64+e] = rmsnorm(nodes[bi]) @ Wr + bias[e]
//   colc [bi*64+e] = rmsnorm(nodes[bi]) @ Wc
// ---------------------------------------------------------------------------
__global__ __launch_bounds__(128) void node_proj_kernel(
    const float* __restrict__ nodes, const float* __restrict__ g_node,
    const float* __restrict__ W, const float* __restrict__ bias,
    float* __restrict__ rowcb, float* __restrict__ colc)
{
    const int bi = blockIdx.x;        // 0 .. B*N-1
    const int t  = threadIdx.x;       // 0 .. 127
    __shared__ float s_red[DIM_NODES];
    __shared__ float s_xn[DIM_NODES];

    float x = nodes[(size_t)bi * DIM_NODES + t];
    s_red[t] = x * x;
    __syncthreads();
    #pragma unroll
    for (int off = 64; off > 0; off >>= 1) {
        if (t < off) s_red[t] += s_red[t + off];
        __syncthreads();
    }
    float scale = rsqrtf(s_red[0] * (1.0f / DIM_NODES) + EPS_F32);
    s_xn[t] = x * scale * g_node[t];
    __syncthreads();

    const int e = t & 63;
    const float* Wp = (t < 64) ? W : (W + DIM_NODES * DIM_EDGES);
    float acc = 0.0f;
    #pragma unroll 8
    for (int d = 0; d < DIM_NODES; ++d)
        acc += s_xn[d] * Wp[d * DIM_EDGES + e];

    if (t < 64) rowcb[(size_t)bi * DIM_EDGES + e] = acc + bias[e];
    else        colc [(size_t)bi * DIM_EDGES + e] = acc;
}

// ---------------------------------------------------------------------------
// Kernel 2: edges RMS-norm + 64x64 projection via V_WMMA_F32_16X16X4_F32,
// fused with row/col broadcast add.
//
// Block = 128 threads (4 wave32s), one (b,i) row per block (512 j's).
// Wave w handles tiles j0 = (it*4 + w)*16, it = 0..7 (8 tiles per wave).
// Async path: double-buffered GLOBAL_LOAD_ASYNC_TO_LDS_B128 prefetches tile
// it+1 while tile it runs its 64 WMMAs; per-row RMS scale folded into the A
// fragments in registers (g_edge folded into B).
// ---------------------------------------------------------------------------
#define WT_STRIDE   68
#define TILE_STRIDE 68
#define NWAVES      4
#define TILES_PER_WAVE 8

__global__ __launch_bounds__(128) void edge_fuse_kernel(
    const float* __restrict__ edges, const float* __restrict__ g_edge,
    const float* __restrict__ W,
    const float* __restrict__ rowcb, const float* __restrict__ colc,
    float* __restrict__ out)
{
    __shared__ float s_Wt[DIM_EDGES * WT_STRIDE];              // 17408 B
#if USE_ASYNC_LDS
    __shared__ float s_tile[NWAVES][2][16 * TILE_STRIDE];      // 34816 B
#else
    __shared__ float s_tile[NWAVES][1][16 * TILE_STRIDE];      // 17408 B
#endif

    const int tid  = threadIdx.x;
    const int lane = tid & 31;
    const int wave = tid >> 5;                 // 0..3
    const int bi   = blockIdx.x;               // b*N + i
    const int b    = bi >> 9;                  // N = 512

    // ---- cooperative load of g_edge-scaled, transposed We into LDS ----
    // s_Wt[e][k] = g_edge[k] * We[k][e]; stride 68 -> conflict-free b64 frags
    const float* We = W + 2 * DIM_NODES * DIM_EDGES;
    #pragma unroll
    for (int itw = 0; itw < 32; ++itw) {
        int idx = itw * 128 + tid;             // 0..4095, coalesced global read
        int k = idx >> 6, e = idx & 63;
        s_Wt[e * WT_STRIDE + k] = g_edge[k] * We[k * DIM_EDGES + e];
    }
    __syncthreads();

    const int half = lane >> 4;                // 0: K%4 in {0,1}, 1: {2,3}
    const int ln   = lane & 15;                // A row / B column lane
    const int row  = lane >> 1;                // tile row for loads
    const int col0 = (lane & 1) * 32;          // half-row offset

    const float* ebase = edges + ((size_t)bi * NPTS) * DIM_EDGES;
    const float* rb    = rowcb + (size_t)bi * DIM_EDGES;
    const float* cc    = colc + ((size_t)b * NPTS) * DIM_EDGES;
    float*       obase = out + ((size_t)bi * NPTS) * DIM_EDGES;

#if USE_ASYNC_LDS
    {   // prefetch tile for it = 0
        const int j0 = wave * 16;
        const float* g = ebase + (size_t)(j0 + row) * DIM_EDGES + col0;
        ASYNC_TILE_LOAD(to_glb(g), to_lds(&s_tile[wave][0][row * TILE_STRIDE + col0]));
    }
#endif

    #pragma unroll 1
    for (int it = 0; it < TILES_PER_WAVE; ++it) {
        const int j0 = (it * NWAVES + wave) * 16;
        v2f afrag[16];

#if USE_ASYNC_LDS
        const float* s_t = &s_tile[wave][it & 1][0];
        if (it + 1 < TILES_PER_WAVE) {
            const int jn = ((it + 1) * NWAVES + wave) * 16;
            const float* g = ebase + (size_t)(jn + row) * DIM_EDGES + col0;
            ASYNC_TILE_LOAD(to_glb(g),
                            to_lds(&s_tile[wave][(it + 1) & 1][row * TILE_STRIDE + col0]));
            WAIT_ASYNC(8);   // in-order: first 8 (tile it) complete
        } else {
            WAIT_ASYNC(0);
        }
        asm volatile("" ::: "memory");

        // read raw A fragments, RMS-normalize in registers
        const float* Arow = s_t + ln * TILE_STRIDE + half * 2;
        float ss = 0.0f;
        #pragma unroll
        for (int kc = 0; kc < 16; ++kc) {
            afrag[kc] = *(const v2f*)(Arow + kc * 4);
            ss += afrag[kc][0] * afrag[kc][0] + afrag[kc][1] * afrag[kc][1];
        }
        // lane L holds K%4={0,1}, lane L^16 holds {2,3} of the same row: SWAPX16
        ss += __int_as_float(__builtin_amdgcn_ds_swizzle(__float_as_int(ss), 0x401f));
        const float scale = rsqrtf(ss * (1.0f / DIM_EDGES) + EPS_F32);
        #pragma unroll
        for (int kc = 0; kc < 16; ++kc) {
            afrag[kc][0] *= scale;
            afrag[kc][1] *= scale;
        }
#else
        // synchronous fallback: load+normalize through VGPRs into LDS
        float* s_t = &s_tile[wave][0][0];
        {
            const float4* src = (const float4*)(ebase + (size_t)(j0 + row) * DIM_EDGES + col0);
            float4 vv[8];
            float ss = 0.0f;
            #pragma unroll
            for (int q = 0; q < 8; ++q) {
                vv[q] = src[q];
                ss += vv[q].x * vv[q].x + vv[q].y * vv[q].y +
                      vv[q].z * vv[q].z + vv[q].w * vv[q].w;
            }
            ss += __int_as_float(__builtin_amdgcn_ds_swizzle(__float_as_int(ss), 0x041f));
            float scale = rsqrtf(ss * (1.0f / DIM_EDGES) + EPS_F32);
            float* dst = s_t + row * TILE_STRIDE + col0;
            #pragma unroll
            for (int q = 0; q < 8; ++q) {
                dst[4 * q + 0] = vv[q].x * scale;
                dst[4 * q + 1] = vv[q].y * scale;
                dst[4 * q + 2] = vv[q].z * scale;
                dst[4 * q + 3] = vv[q].w * scale;
            }
        }
        const float* Arow = s_t + ln * TILE_STRIDE + half * 2;
        #pragma unroll
        for (int kc = 0; kc < 16; ++kc)
            afrag[kc] = *(const v2f*)(Arow + kc * 4);
#endif

        // ---- accumulators init with rowcb (row_proj + bias), constant over M
        v8f acc[4];
        #pragma unroll
        for (int nt = 0; nt < 4; ++nt) {
            float r = rb[nt * 16 + ln];
            #pragma unroll
            for (int v = 0; v < 8; ++v) acc[nt][v] = r;
        }

        // ---- GEMM: 16 K-chunks x 4 N-tiles of v_wmma_f32_16x16x4_f32
        #pragma unroll
        for (int kc = 0; kc < 16; ++kc) {
            #pragma unroll
            for (int nt = 0; nt < 4; ++nt) {
                v2f bf = *(const v2f*)(s_Wt + (nt * 16 + ln) * WT_STRIDE + kc * 4 + half * 2);
                acc[nt] = __builtin_amdgcn_wmma_f32_16x16x4_f32(
                    /*neg_a=*/false, afrag[kc], /*neg_b=*/false, bf,
                    /*c_mod=*/(short)0, acc[nt],
                    /*reuse_a=*/false, /*reuse_b=*/false);
            }
        }

        // ---- epilogue: add colc[j, e], NT-store the output stream ----
        float* optr = obase + (size_t)j0 * DIM_EDGES;
        #pragma unroll
        for (int nt = 0; nt < 4; ++nt) {
            const int e = nt * 16 + ln;
            #pragma unroll
            for (int v = 0; v < 8; ++v) {
                const int jr = half * 8 + v;   // C layout: M = v + 8*half
                float val = acc[nt][v] + cc[(size_t)(j0 + jr) * DIM_EDGES + e];
                __builtin_nontemporal_store(val, optr + (size_t)jr * DIM_EDGES + e);
            }
        }
    }
}

// ---------------------------------------------------------------------------
extern "C" void kernel_launch(void* const* d_in, const int* in_sizes, int n_in,
                              void* d_out, int out_size, void* d_ws, size_t ws_size,
                              hipStream_t stream) {
    const float* edges  = (const float*)d_in[0];
    const float* nodes  = (const float*)d_in[1];
    const float* g_node = (const float*)d_in[2];
    const float* g_edge = (const float*)d_in[3];
    const float* W      = (const float*)d_in[4];
    const float* bias   = (const float*)d_in[5];
    float* out = (float*)d_out;

    float* rowcb = (float*)d_ws;                               // B*N*64 floats
    float* colc  = rowcb + (size_t)BATCH * NPTS * DIM_EDGES;   // B*N*64 floats

    node_proj_kernel<<<dim3(BATCH * NPTS), dim3(128), 0, stream>>>(
        nodes, g_node, W, bias, rowcb, colc);

    edge_fuse_kernel<<<dim3(BATCH * NPTS), dim3(128), 0, stream>>>(
        edges, g_edge, W, rowcb, colc, out);
}